// RGBICPPoseEstimator_22720376996610
// MI455X (gfx1250) — compile-verified
//
#include <hip/hip_runtime.h>
#include <math.h>

#define IMG_H 480
#define IMG_W 640
#define NPIX (IMG_H * IMG_W)
#define NCHUNK (NPIX / 32)        // 9600; 640 = 20*32 so each chunk is within one row
#define CHUNKS_PER_ROW (IMG_W / 32)
#define NITER 20

#define ICP_WT 10.0f
#define SQRT_ICP_WT 3.16227766016838f
#define FTOL 1e-3f
#define XTOL 1e-8f
#define DIST_THR (200.0f / 15.0f)
#define NORMAL_THR 0.94f

typedef __attribute__((ext_vector_type(2))) float v2f;
typedef __attribute__((ext_vector_type(8))) float v8f;

// workspace layout (floats)
#define WS_RED  0     // 256-entry 16x16 Gram accumulator
#define WS_X    256   // 6
#define WS_CONV 262
#define WS_COST 263
#define WS_R    264   // 9, row-major
#define WS_T    273   // 3

static __device__ __forceinline__ float clampf(float v, float lo, float hi) {
    return fminf(fmaxf(v, lo), hi);
}

__device__ void exp_se3_dev(const float xi[6], float R[9], float t[3]) {
    float vx = xi[0], vy = xi[1], vz = xi[2];
    float wx = xi[3], wy = xi[4], wz = xi[5];
    float th2 = wx * wx + wy * wy + wz * wz;
    float th  = sqrtf(th2);
    bool small = th < 1e-8f;
    float ths = small ? 1.0f : th;
    float A = small ? (1.0f - th2 / 6.0f)          : (sinf(ths) / ths);
    float B = small ? (0.5f - th2 / 24.0f)         : ((1.0f - cosf(ths)) / (ths * ths));
    float C = small ? (1.0f / 6.0f - th2 / 120.0f) : ((ths - sinf(ths)) / (ths * ths * ths));
    float Wm[9] = {0.f, -wz, wy,   wz, 0.f, -wx,   -wy, wx, 0.f};
    float W2[9];
    for (int i = 0; i < 3; ++i)
        for (int j = 0; j < 3; ++j) {
            float s = 0.f;
            for (int k = 0; k < 3; ++k) s += Wm[i * 3 + k] * Wm[k * 3 + j];
            W2[i * 3 + j] = s;
        }
    float V[9];
    for (int i = 0; i < 9; ++i) {
        float I = (i == 0 || i == 4 || i == 8) ? 1.f : 0.f;
        R[i] = I + A * Wm[i] + B * W2[i];
        V[i] = I + B * Wm[i] + C * W2[i];
    }
    t[0] = V[0] * vx + V[1] * vy + V[2] * vz;
    t[1] = V[3] * vx + V[4] * vy + V[5] * vz;
    t[2] = V[6] * vx + V[7] * vy + V[8] * vz;
}

__global__ void icp_init(float* ws) {
    int tid = threadIdx.x;
    ws[WS_RED + tid] = 0.f;
    if (tid == 0) {
        for (int i = 0; i < 6; ++i) ws[WS_X + i] = 0.f;
        ws[WS_CONV] = 0.f;
        ws[WS_COST] = 0.f;
        for (int i = 0; i < 9; ++i) ws[WS_R + i] = (i % 4 == 0) ? 1.f : 0.f;
        ws[WS_T + 0] = 0.f; ws[WS_T + 1] = 0.f; ws[WS_T + 2] = 0.f;
    }
}

// Per-pixel residual/Jacobian pass + WMMA Gram accumulation.
__global__ __launch_bounds__(256) void icp_reduce(
    const float* __restrict__ depth_ref,
    const float* __restrict__ normals_ref,
    const float* __restrict__ gray_ref,
    const float* __restrict__ gray_target,
    const float* __restrict__ points_target,
    const float* __restrict__ normals_target,
    const float* __restrict__ intr,
    float* __restrict__ ws)
{
    __shared__ float xpose[8][32 * 17];   // per-wave transpose staging (stride 17 vs bank conflicts)
    __shared__ float redl[256];

    const int tid  = threadIdx.x;
    // wave index is wave-uniform: pin it to an SGPR so all chunk/row math scalarizes
    const int wave = __builtin_amdgcn_readfirstlane(tid >> 5);
    const int lane = tid & 31;

    redl[tid] = 0.f;

    // broadcast pose + intrinsics (uniform, L2-cached)
    const float R00 = ws[WS_R + 0], R01 = ws[WS_R + 1], R02 = ws[WS_R + 2];
    const float R10 = ws[WS_R + 3], R11 = ws[WS_R + 4], R12 = ws[WS_R + 5];
    const float R20 = ws[WS_R + 6], R21 = ws[WS_R + 7], R22 = ws[WS_R + 8];
    const float t0 = ws[WS_T + 0], t1 = ws[WS_T + 1], t2 = ws[WS_T + 2];
    const float fx = intr[0], fy = intr[4], cx = intr[2], cy = intr[5];
    const float inv_fx = 1.0f / fx;       // hoisted uniform reciprocals
    const float inv_fy = 1.0f / fy;

    v8f c0 = {0.f, 0.f, 0.f, 0.f, 0.f, 0.f, 0.f, 0.f};
    v8f c1 = {0.f, 0.f, 0.f, 0.f, 0.f, 0.f, 0.f, 0.f};

    const int gwave  = blockIdx.x * 8 + wave;     // scalar
    const int nwaves = gridDim.x * 8;             // scalar
    const float* __restrict__ g = gray_target;

    for (int chunk = gwave; chunk < NCHUNK; chunk += nwaves) {   // scalar loop counter
        const int row  = chunk / CHUNKS_PER_ROW;                  // SALU (uniform)
        const int col0 = (chunk - row * CHUNKS_PER_ROW) * 32;     // SALU (uniform)
        const int px   = col0 + lane;
        const int py   = row;
        const int p    = row * IMG_W + px;

        // back-project reference pixel
        const float Z  = depth_ref[p];
        const float Px = ((float)px - cx) * inv_fx * Z;
        const float Py = ((float)py - cy) * inv_fy * Z;
        const float Pz = Z;
        const float Nrx = normals_ref[3 * p + 0];
        const float Nry = normals_ref[3 * p + 1];
        const float Nrz = normals_ref[3 * p + 2];
        const float Ir  = gray_ref[p];

        // transform
        const float Ptx = R00 * Px + R01 * Py + R02 * Pz + t0;
        const float Pty = R10 * Px + R11 * Py + R12 * Pz + t1;
        const float Ptz = R20 * Px + R21 * Py + R22 * Pz + t2;
        const bool okz = Ptz > 1e-6f;
        const float Zs = okz ? Ptz : 1.0f;
        const float rz = 1.0f / Zs;                               // single division per pixel
        const float fxPtx = fx * Ptx, fyPty = fy * Pty;
        const float uu = fxPtx * rz + cx;
        const float vv = fyPty * rz + cy;
        const bool inb = okz && (uu >= 0.f) && (uu <= (float)(IMG_W - 1)) &&
                                (vv >= 0.f) && (vv <= (float)(IMG_H - 1));

        // nearest-neighbor association (round half-even like jnp.round)
        const int ui = (int)clampf(rintf(uu), 0.f, (float)(IMG_W - 1));
        const int vi = (int)clampf(rintf(vv), 0.f, (float)(IMG_H - 1));
        const int qi = (vi * IMG_W + ui) * 3;
        const float qx = points_target[qi + 0], qy = points_target[qi + 1], qz = points_target[qi + 2];
        const float nqx = normals_target[qi + 0], nqy = normals_target[qi + 1], nqz = normals_target[qi + 2];
        const float dx = Ptx - qx, dy = Pty - qy, dz = Ptz - qz;

        const float nrx = R00 * Nrx + R01 * Nry + R02 * Nrz;
        const float nry = R10 * Nrx + R11 * Nry + R12 * Nrz;
        const float nrz = R20 * Nrx + R21 * Nry + R22 * Nrz;

        const bool vicp = inb && (dx * dx + dy * dy + dz * dz < DIST_THR * DIST_THR) &&
                                 (nrx * nqx + nry * nqy + nrz * nqz > NORMAL_THR);
        const float mi = vicp ? 1.f : 0.f;
        const float mr = inb ? 1.f : 0.f;

        const float ricp = (nqx * dx + nqy * dy + nqz * dz) * mi;
        const float cix = Pty * nqz - Ptz * nqy;
        const float ciy = Ptz * nqx - Ptx * nqz;
        const float ciz = Ptx * nqy - Pty * nqx;

        // bilinear footprint
        const float u0f = floorf(uu), v0f = floorf(vv);
        const float du = uu - u0f, dv = vv - v0f;
        const int u0 = (int)clampf(u0f, 0.f, (float)(IMG_W - 1));
        const int u1 = min(u0 + 1, IMG_W - 1);
        const int v0 = (int)clampf(v0f, 0.f, (float)(IMG_H - 1));
        const int v1 = min(v0 + 1, IMG_H - 1);

        // branchless jnp.gradient samples (one-sided at edges via clamp + scale select)
        auto gx_at = [&](int y, int x) -> float {
            const int xm = max(x - 1, 0);
            const int xp = min(x + 1, IMG_W - 1);
            const float s = (xp - xm == 2) ? 0.5f : 1.0f;
            return (g[y * IMG_W + xp] - g[y * IMG_W + xm]) * s;
        };
        auto gy_at = [&](int y, int x) -> float {
            const int ym = max(y - 1, 0);
            const int yp = min(y + 1, IMG_H - 1);
            const float s = (yp - ym == 2) ? 0.5f : 1.0f;
            return (g[yp * IMG_W + x] - g[ym * IMG_W + x]) * s;
        };
        auto bil = [&](float a00, float a01, float a10, float a11) -> float {
            return (a00 * (1.f - du) + a01 * du) * (1.f - dv) + (a10 * (1.f - du) + a11 * du) * dv;
        };

        const float It  = bil(g[v0 * IMG_W + u0], g[v0 * IMG_W + u1],
                              g[v1 * IMG_W + u0], g[v1 * IMG_W + u1]);
        const float gxs = bil(gx_at(v0, u0), gx_at(v0, u1), gx_at(v1, u0), gx_at(v1, u1));
        const float gys = bil(gy_at(v0, u0), gy_at(v0, u1), gy_at(v1, u0), gy_at(v1, u1));

        const float rrgb = (It - Ir) * mr;
        const float gxf = gxs * fx, gyf = gys * fy;
        const float gpx = gxf * rz;
        const float gpy = gyf * rz;
        const float gpz = -(gxf * Ptx + gyf * Pty) * (rz * rz);
        const float cgx = Pty * gpz - Ptz * gpy;
        const float cgy = Ptz * gpx - Ptx * gpz;
        const float cgz = Ptx * gpy - Pty * gpx;

        // 16-vector: [sqrt(w)*J_icp, sqrt(w)*r_icp, 0, J_rgb, r_rgb, 0]
        float j[16];
        j[0] = SQRT_ICP_WT * nqx * mi; j[1] = SQRT_ICP_WT * nqy * mi; j[2] = SQRT_ICP_WT * nqz * mi;
        j[3] = SQRT_ICP_WT * cix * mi; j[4] = SQRT_ICP_WT * ciy * mi; j[5] = SQRT_ICP_WT * ciz * mi;
        j[6] = SQRT_ICP_WT * ricp;     j[7] = 0.f;
        j[8]  = gpx * mr; j[9]  = gpy * mr; j[10] = gpz * mr;
        j[11] = cgx * mr; j[12] = cgy * mr; j[13] = cgz * mr;
        j[14] = rrgb;     j[15] = 0.f;

        // transpose through per-wave LDS region (intra-wave only; DS ops of a
        // wave complete in order — force the store->load order with dscnt wait)
        float* rowp = &xpose[wave][lane * 17];
        #pragma unroll
        for (int e = 0; e < 16; ++e) rowp[e] = j[e];
        asm volatile("s_wait_dscnt 0" ::: "memory");

        const float* base = &xpose[wave][0];
        const int ii = lane & 15;
        const int h2 = (lane >> 4) * 2;
        // 8 WMMAs cover 32 pixels: D += A(16x4) * A^T, A cols = 4 pixel-vectors.
        // wave32 A(16x4)/B(4x16) f32 operand layouts coincide -> pass same regs.
        #pragma unroll
        for (int w = 0; w < 8; w += 2) {
            v2f a;
            a.x = base[(4 * w + h2 + 0) * 17 + ii];
            a.y = base[(4 * w + h2 + 1) * 17 + ii];
            c0 = __builtin_amdgcn_wmma_f32_16x16x4_f32(false, a, false, a, (short)0, c0, false, false);
            v2f b;
            b.x = base[(4 * (w + 1) + h2 + 0) * 17 + ii];
            b.y = base[(4 * (w + 1) + h2 + 1) * 17 + ii];
            c1 = __builtin_amdgcn_wmma_f32_16x16x4_f32(false, b, false, b, (short)0, c1, false, false);
        }
    }

    // combine accumulators, reduce across waves via LDS, then global atomics
    #pragma unroll
    for (int r = 0; r < 8; ++r) c0[r] += c1[r];

    __syncthreads();
    const int N = lane & 15;
    const int mhi = (lane >> 4) * 8;
    #pragma unroll
    for (int r = 0; r < 8; ++r) {
        const int M = r + mhi;
        atomicAdd(&redl[M * 16 + N], c0[r]);
    }
    __syncthreads();
    atomicAdd(&ws[WS_RED + tid], redl[tid]);
}

// 6x6 solve + convergence logic + pose update for next iteration.
__global__ void icp_solve(float* ws) {
    float M[6][7];
    for (int i = 0; i < 6; ++i) {
        for (int j = 0; j < 6; ++j)
            M[i][j] = ws[i * 16 + j] + ws[(8 + i) * 16 + (8 + j)];
        M[i][6] = ws[i * 16 + 6] + ws[(8 + i) * 16 + 14];
        M[i][i] += 1e-9f;
    }
    const float sumsq = ws[6 * 16 + 6] + ws[14 * 16 + 14];
    const float new_cost = sumsq / (float)NPIX;

    for (int k = 0; k < 256; ++k) ws[WS_RED + k] = 0.f;   // re-arm accumulator

    // Gaussian elimination with partial pivoting
    for (int k = 0; k < 6; ++k) {
        int piv = k; float best = fabsf(M[k][k]);
        for (int r = k + 1; r < 6; ++r) {
            float a = fabsf(M[r][k]);
            if (a > best) { best = a; piv = r; }
        }
        if (piv != k)
            for (int c = k; c < 7; ++c) { float tmp = M[k][c]; M[k][c] = M[piv][c]; M[piv][c] = tmp; }
        const float inv = 1.f / M[k][k];
        for (int r = k + 1; r < 6; ++r) {
            const float f = M[r][k] * inv;
            for (int c = k; c < 7; ++c) M[r][c] -= f * M[k][c];
        }
    }
    float x0[6];
    for (int i = 5; i >= 0; --i) {
        float s = M[i][6];
        for (int j = i + 1; j < 6; ++j) s -= M[i][j] * x0[j];
        x0[i] = s / M[i][i];
    }

    float x[6];
    float n0 = 0.f, nx = 0.f;
    for (int i = 0; i < 6; ++i) { x[i] = ws[WS_X + i]; n0 += x0[i] * x0[i]; nx += x[i] * x[i]; }
    n0 = sqrtf(n0); nx = sqrtf(nx);
    const bool conv_old = ws[WS_CONV] != 0.f;
    const float cost_old = ws[WS_COST];
    const bool c = conv_old || (new_cost < FTOL) || (n0 < XTOL * (XTOL + nx));

    float xn[6];
    for (int i = 0; i < 6; ++i) xn[i] = c ? x[i] : (x[i] - x0[i]);
    for (int i = 0; i < 6; ++i) ws[WS_X + i] = xn[i];
    ws[WS_CONV] = c ? 1.f : 0.f;
    ws[WS_COST] = conv_old ? cost_old : new_cost;

    float R[9], t[3];
    exp_se3_dev(xn, R, t);
    for (int i = 0; i < 9; ++i) ws[WS_R + i] = R[i];
    for (int i = 0; i < 3; ++i) ws[WS_T + i] = t[i];
}

__global__ void icp_final(const float* ws, float* out) {
    float x[6];
    for (int i = 0; i < 6; ++i) x[i] = ws[WS_X + i];
    float R[9], t[3];
    exp_se3_dev(x, R, t);
    out[0] = R[0]; out[1] = R[1]; out[2]  = R[2];  out[3]  = t[0];
    out[4] = R[3]; out[5] = R[4]; out[6]  = R[5];  out[7]  = t[1];
    out[8] = R[6]; out[9] = R[7]; out[10] = R[8];  out[11] = t[2];
    out[12] = 0.f; out[13] = 0.f; out[14] = 0.f;   out[15] = 1.f;
    out[16] = ws[WS_COST];
}

extern "C" void kernel_launch(void* const* d_in, const int* in_sizes, int n_in,
                              void* d_out, int out_size, void* d_ws, size_t ws_size,
                              hipStream_t stream) {
    (void)in_sizes; (void)n_in; (void)out_size; (void)ws_size;
    const float* depth_ref      = (const float*)d_in[0];
    const float* normals_ref    = (const float*)d_in[1];
    const float* gray_ref       = (const float*)d_in[2];
    const float* gray_target    = (const float*)d_in[3];
    const float* points_target  = (const float*)d_in[4];
    const float* normals_target = (const float*)d_in[5];
    const float* intrinsics     = (const float*)d_in[6];
    float* ws  = (float*)d_ws;
    float* out = (float*)d_out;

    icp_init<<<1, 256, 0, stream>>>(ws);
    for (int it = 0; it < NITER; ++it) {
        // 400 blocks * 8 waves = 3200 waves -> exactly 3 chunks (96 pixels) per wave
        icp_reduce<<<400, 256, 0, stream>>>(depth_ref, normals_ref, gray_ref, gray_target,
                                            points_target, normals_target, intrinsics, ws);
        icp_solve<<<1, 1, 0, stream>>>(ws);
    }
    icp_final<<<1, 1, 0, stream>>>(ws, out);
}